// DistillLoss_88476326298380
// MI455X (gfx1250) — compile-verified
//
#include <hip/hip_runtime.h>
#include <hip/hip_bf16.h>

typedef __attribute__((ext_vector_type(2))) float v2f;
typedef __attribute__((ext_vector_type(8))) float v8f;

#define BLOCKN 256
#define CHUNK 64
#define LSTRIDE 68   // CHUNK + 4; stride % 64 == 4 -> conflict-free A-frag b64 reads
#define MMAX 128

// Async global->LDS copy via inline asm (bypasses the clang builtin, whose
// signature differs across toolchains). VDST = per-lane LDS byte address,
// VADDR = 64-bit global address, saddr = off (GV mode).
__device__ __forceinline__ void async_load_lds_b64(const float* src, float* lds_dst) {
    const unsigned lds_off =
        (unsigned)(size_t)(__attribute__((address_space(3))) float*)lds_dst;
    const unsigned long long gaddr = (unsigned long long)src;
    asm volatile("global_load_async_to_lds_b64 %0, %1, off"
                 :
                 : "v"(lds_off), "v"(gaddr)
                 : "memory");
}

__device__ __forceinline__ void wait_async_lds() {
#if __has_builtin(__builtin_amdgcn_s_wait_asynccnt)
    __builtin_amdgcn_s_wait_asynccnt(0);
#else
    asm volatile("s_wait_asynccnt 0x0" ::: "memory");
#endif
}

__global__ __launch_bounds__(BLOCKN) void distill_main_kernel(
    const float* __restrict__ qe,
    const float* __restrict__ de,
    const float* __restrict__ soft,
    const int*   __restrict__ ndarr,
    float* __restrict__ partials,
    int Bn, int Dn, int Mn)
{
    __shared__ float tile[2][MMAX * LSTRIDE];   // double-buffered doc tile
    __shared__ float q_s[1024];
    __shared__ float sim_s[MMAX];
    __shared__ float redf[BLOCKN];
    __shared__ int   redi[BLOCKN];

    const int t    = threadIdx.x;
    const int b    = blockIdx.x;
    const int lane = t & 31;        // wave32
    const int wave = t >> 5;

    // ---- ragged segment offset for this sample: sum(nd[0..b-1]) ----
    int acc = 0;
    for (int i = t; i < Bn; i += BLOCKN) {
        int v = ndarr[i];
        acc += (i < b) ? v : 0;
    }
    redi[t] = acc;
    __syncthreads();
    for (int s = BLOCKN >> 1; s > 0; s >>= 1) {
        if (t < s) redi[t] += redi[t + s];
        __syncthreads();
    }
    const int offset = redi[0];
    const int nd     = ndarr[b];

    // ---- query row into LDS (made visible by the prologue barrier below) ----
    for (int i = t; i < Dn; i += BLOCKN) q_s[i] = qe[(size_t)b * Dn + i];

    // ---- WMMA GEMV: sim[m] = dot(doc[offset+m], q) ----
    // wave w owns rows m0..m0+15; A frag: lane L<16 -> row m0+L, K{0,1}; lanes 16-31 -> K{2,3}
    const int m0 = wave * 16;
    const int hf = lane >> 4;
    const int lq = lane & 15;
    const float* docbase = de + (size_t)offset * (size_t)Dn;

    // rows this wave actually needs to fetch (uniform scalar trip count)
    int rmax = nd - m0;
    rmax = rmax < 0 ? 0 : (rmax > 16 ? 16 : rmax);

    // stage rows m0..m0+rmax-1, cols [c0, c0+CHUNK) into tile[buf]; 8B/lane coalesced
    auto stage = [&](int c0, int buf) {
        for (int r = 0; r < rmax; ++r) {
            const int m = m0 + r;
            const float* src = docbase + (size_t)m * Dn + c0 + 2 * lane;
            float* dst = &tile[buf][m * LSTRIDE + 2 * lane];
            async_load_lds_b64(src, dst);
        }
    };

    v8f cacc = {};
    const int NCH = Dn / CHUNK;

    // prologue: chunk 0 into buffer 0
    stage(0, 0);
    wait_async_lds();
    __syncthreads();

    for (int ci = 0; ci < NCH; ++ci) {
        const int cur = ci & 1;
        if (ci + 1 < NCH) stage((ci + 1) * CHUNK, cur ^ 1);

        const float* arow = &tile[cur][(m0 + lq) * LSTRIDE + 2 * hf];
        const float* qrow = &q_s[ci * CHUNK + 2 * hf];
#pragma unroll
        for (int kk = 0; kk < CHUNK; kk += 4) {
            v2f a  = *(const v2f*)(arow + kk);         // A: doc rows tile
            v2f bb = *(const v2f*)(qrow + kk);         // B: q chunk broadcast over N
            cacc = __builtin_amdgcn_wmma_f32_16x16x4_f32(
                false, a, false, bb, 0, cacc, false, false);
        }
        wait_async_lds();   // next chunk's async copies have landed
        __syncthreads();    // everyone done reading tile[cur] before it is rewritten
    }

    // extract column 0 of C: lane 0 holds rows m0..m0+7, lane 16 holds m0+8..m0+15
    if (lane == 0) {
#pragma unroll
        for (int i = 0; i < 8; ++i) sim_s[m0 + i] = cacc[i];
    } else if (lane == 16) {
#pragma unroll
        for (int i = 0; i < 8; ++i) sim_s[m0 + 8 + i] = cacc[i];
    }
    __syncthreads();

    // ---- masked log-softmax + teacher-normalized KL ----
    const float NEGV = -1e30f;
    const float invT = 50.0f;   // 1 / 0.02
    float myv = NEGV;
    if (t < Mn) myv = (t < nd) ? sim_s[t] * invT : NEGV;

    // max
    redf[t] = (t < Mn) ? myv : NEGV;
    __syncthreads();
    for (int s = BLOCKN >> 1; s > 0; s >>= 1) {
        if (t < s) redf[t] = fmaxf(redf[t], redf[t + s]);
        __syncthreads();
    }
    const float mx = redf[0];
    __syncthreads();

    // sum of exp
    const float ex = (t < Mn) ? expf(myv - mx) : 0.0f;
    redf[t] = ex;
    __syncthreads();
    for (int s = BLOCKN >> 1; s > 0; s >>= 1) {
        if (t < s) redf[t] += redf[t + s];
        __syncthreads();
    }
    const float Z = redf[0];
    __syncthreads();
    const float logZ = logf(Z);

    // teacher normalization
    const float traw = (t < Mn && t < nd) ? soft[(size_t)b * Mn + t] : 0.0f;
    redf[t] = traw;
    __syncthreads();
    for (int s = BLOCKN >> 1; s > 0; s >>= 1) {
        if (t < s) redf[t] += redf[t + s];
        __syncthreads();
    }
    const float S = redf[0];
    __syncthreads();

    float kl = 0.0f;
    if (traw > 0.0f) {
        const float tn   = traw / (S + 1e-9f);
        const float logp = myv - mx - logZ;
        kl = tn * (logf(tn) - logp);
    }
    redf[t] = kl;
    __syncthreads();
    for (int s = BLOCKN >> 1; s > 0; s >>= 1) {
        if (t < s) redf[t] += redf[t + s];
        __syncthreads();
    }
    if (t == 0) partials[b] = redf[0];
}

// deterministic final reduction: partials[0..B) -> out[0] = sum / B
__global__ __launch_bounds__(BLOCKN) void distill_final_kernel(
    const float* __restrict__ partials, float* __restrict__ out, int Bn)
{
    __shared__ float redf[BLOCKN];
    const int t = threadIdx.x;
    float acc = 0.0f;
    for (int i = t; i < Bn; i += BLOCKN) acc += partials[i];
    redf[t] = acc;
    __syncthreads();
    for (int s = BLOCKN >> 1; s > 0; s >>= 1) {
        if (t < s) redf[t] += redf[t + s];
        __syncthreads();
    }
    if (t == 0) out[0] = redf[0] / (float)Bn;
}

extern "C" void kernel_launch(void* const* d_in, const int* in_sizes, int n_in,
                              void* d_out, int out_size, void* d_ws, size_t ws_size,
                              hipStream_t stream) {
    const float* qe    = (const float*)d_in[0];   // [B, D] f32
    const float* de    = (const float*)d_in[1];   // [total, D] f32
    const float* soft  = (const float*)d_in[2];   // [B, M] f32
    const int*   ndarr = (const int*)d_in[3];     // [B] i32

    const int Bn = in_sizes[3];
    const int Dn = in_sizes[0] / Bn;
    const int Mn = in_sizes[2] / Bn;

    float* partials = (float*)d_ws;               // Bn floats of scratch
    float* out      = (float*)d_out;

    distill_main_kernel<<<Bn, BLOCKN, 0, stream>>>(qe, de, soft, ndarr, partials,
                                                   Bn, Dn, Mn);
    distill_final_kernel<<<1, BLOCKN, 0, stream>>>(partials, out, Bn);
}